// MultiHeadAttention_24867860644365
// MI455X (gfx1250) — compile-verified
//
#include <hip/hip_runtime.h>
#include <math.h>
#include <stdint.h>

// Problem constants (from reference): B=2, S=2048, D=1024, H=16, dk=64
#define BB 2
#define SS 2048
#define DD 1024
#define HH 16
#define DK 64
#define LDT 68  // padded LDS row stride (floats): frag b64 reads hit banks (4m+k)%64, conflict-free
#define LDV 80  // pad for V staged [k][n]: b32 frag reads hit banks (16k+n)%64, conflict-free

#ifndef __has_builtin
#define __has_builtin(x) 0
#endif
#if __has_builtin(__builtin_amdgcn_global_load_async_to_lds_b128) && \
    __has_builtin(__builtin_amdgcn_s_wait_asynccnt)
#define USE_ASYNC 1
#else
#define USE_ASYNC 0
#endif

typedef __attribute__((ext_vector_type(2))) float v2f;
typedef __attribute__((ext_vector_type(8))) float v8f;
// Pointee type of the async builtin's params (per hipcc diagnostic):
// '__attribute__((__vector_size__(4 * sizeof(int)))) int'
typedef int b128_t __attribute__((vector_size(16)));

// Address-space casts for the async-to-LDS builtin (AS1 = global, AS3 = LDS).
// Generic->AS3 via 32-bit truncation of the flat address (LDS address is
// the low 32 bits); generic->AS1 is bit-identical.
#define GLBPTR(p) ((__attribute__((address_space(1))) b128_t*)(uintptr_t)(p))
#define LDSPTR(p) ((__attribute__((address_space(3))) b128_t*)(uint32_t)(uintptr_t)(p))

template <int N>
__device__ __forceinline__ void wait_async() {
#if USE_ASYNC
  __builtin_amdgcn_s_wait_asynccnt(N);
#endif
}

// ---------------------------------------------------------------------------
// WMMA f32 16x16x4: D = A(16x4) * B(4x16) + C(16x16), full fp32 precision.
// ---------------------------------------------------------------------------
__device__ __forceinline__ v8f wmma4(v2f a, v2f b, v8f c) {
  return __builtin_amdgcn_wmma_f32_16x16x4_f32(false, a, false, b, (short)0, c,
                                               false, false);
}

// A/B fragment from an LDS tile stored row-major [row][k], stride LDT.
// Lane L holds {T[row0+(L&15)][k+2*(L>>4)], T[...][k+2*(L>>4)+1]}.
__device__ __forceinline__ v2f frag_ld(const float* t, int row0, int k) {
  const int lane = threadIdx.x & 31;
  const float* p = t + (row0 + (lane & 15)) * LDT + (k + ((lane >> 4) << 1));
  return *reinterpret_cast<const v2f*>(p);  // 8B aligned -> ds_load_b64
}

// B fragment from a tile stored [k][n] with stride LDV (used for V in A*V).
__device__ __forceinline__ v2f frag_ld_kn(const float* t, int n0, int k) {
  const int lane = threadIdx.x & 31;
  const int kk = k + ((lane >> 4) << 1);
  const int n = n0 + (lane & 15);
  v2f r;
  r.x = t[kk * LDV + n];
  r.y = t[(kk + 1) * LDV + n];
  return r;
}

// 64x64 block-tile MMA over one K-chunk of 64 ("NT": both tiles [row][k]).
__device__ __forceinline__ void mma_tile(const float* As, const float* Bs,
                                         int wm, int wn, v8f acc[2][2]) {
#pragma unroll
  for (int k = 0; k < 64; k += 4) {
    v2f a0 = frag_ld(As, wm * 32, k);
    v2f a1 = frag_ld(As, wm * 32 + 16, k);
    v2f b0 = frag_ld(Bs, wn * 32, k);
    v2f b1 = frag_ld(Bs, wn * 32 + 16, k);
    acc[0][0] = wmma4(a0, b0, acc[0][0]);
    acc[0][1] = wmma4(a0, b1, acc[0][1]);
    acc[1][0] = wmma4(a1, b0, acc[1][0]);
    acc[1][1] = wmma4(a1, b1, acc[1][1]);
  }
}

// Same, but B tile is stored [k][n] (natural V layout, async-stageable).
__device__ __forceinline__ void mma_tile_nn(const float* As, const float* Vs,
                                            int wm, int wn, v8f acc[2][2]) {
#pragma unroll
  for (int k = 0; k < 64; k += 4) {
    v2f a0 = frag_ld(As, wm * 32, k);
    v2f a1 = frag_ld(As, wm * 32 + 16, k);
    v2f b0 = frag_ld_kn(Vs, wn * 32, k);
    v2f b1 = frag_ld_kn(Vs, wn * 32 + 16, k);
    acc[0][0] = wmma4(a0, b0, acc[0][0]);
    acc[0][1] = wmma4(a0, b1, acc[0][1]);
    acc[1][0] = wmma4(a1, b0, acc[1][0]);
    acc[1][1] = wmma4(a1, b1, acc[1][1]);
  }
}

// Stage a 64x64 fp32 tile (row-major, row stride ldg) into LDS [64][LDD].
// Async path: 8 global_load_async_to_lds_b128 per wave, tracked by ASYNCcnt.
template <int LDD>
__device__ __forceinline__ void stage64(float* dst, const float* src, int ldg) {
  const int t = threadIdx.x;
  const int col = (t & 15) << 2;
  const int row = t >> 4;
#pragma unroll
  for (int r = 0; r < 64; r += 8) {
#if USE_ASYNC
    __builtin_amdgcn_global_load_async_to_lds_b128(
        GLBPTR(src + (size_t)(row + r) * ldg + col),
        LDSPTR(dst + (row + r) * LDD + col), 0, 0);
#else
    const float4 v =
        *reinterpret_cast<const float4*>(src + (size_t)(row + r) * ldg + col);
    *reinterpret_cast<float4*>(dst + (row + r) * LDD + col) = v;
#endif
  }
}

// ---------------------------------------------------------------------------
// Kernel 1: fused QKV projection.  y = x @ W^T + b for W in {wq,wk,wv}.
// Double-buffered async staging; x tile staged once, reused for 3 weights.
// Output head-major (B,H,S,dk).
// ---------------------------------------------------------------------------
__global__ __launch_bounds__(128) void qkv_proj_kernel(
    const float* __restrict__ x,
    const float* __restrict__ wq, const float* __restrict__ bq,
    const float* __restrict__ wk, const float* __restrict__ bk,
    const float* __restrict__ wv, const float* __restrict__ bv,
    float* __restrict__ Q, float* __restrict__ K, float* __restrict__ V) {
  __shared__ float As[2][64 * LDT];
  __shared__ float Bq[2][64 * LDT];
  __shared__ float Bk[2][64 * LDT];
  __shared__ float Bv[2][64 * LDT];
  const int mb = blockIdx.x;
  const int nb = blockIdx.y;
  const int wave = threadIdx.x >> 5;
  const int wm = wave >> 1, wn = wave & 1;

  v8f zero = {};
  v8f acc[3][2][2];
#pragma unroll
  for (int q = 0; q < 3; ++q)
#pragma unroll
    for (int i = 0; i < 2; ++i)
#pragma unroll
      for (int j = 0; j < 2; ++j) acc[q][i][j] = zero;

  auto issue = [&](int buf, int k0) {
    stage64<LDT>(As[buf], x + (size_t)(mb * 64) * DD + k0, DD);
    stage64<LDT>(Bq[buf], wq + (size_t)(nb * 64) * DD + k0, DD);
    stage64<LDT>(Bk[buf], wk + (size_t)(nb * 64) * DD + k0, DD);
    stage64<LDT>(Bv[buf], wv + (size_t)(nb * 64) * DD + k0, DD);  // 32 issues/wave
  };

  issue(0, 0);
#pragma unroll 1
  for (int c = 0; c < DD / 64; ++c) {
    const int cur = c & 1;
    if (c + 1 < DD / 64) {
      issue(cur ^ 1, (c + 1) * 64);
      wait_async<32>();  // chunk c's 32 loads complete (in-order), c+1 in flight
    } else {
      wait_async<0>();
    }
    __syncthreads();
    mma_tile(As[cur], Bq[cur], wm, wn, acc[0]);
    mma_tile(As[cur], Bk[cur], wm, wn, acc[1]);
    mma_tile(As[cur], Bv[cur], wm, wn, acc[2]);
    __syncthreads();  // all waves done reading before buffer is overwritten
  }

  const int lane = threadIdx.x & 31;
  const int lh = lane >> 4, lc = lane & 15;
#pragma unroll
  for (int i = 0; i < 2; ++i)
#pragma unroll
    for (int j = 0; j < 2; ++j) {
      const int gm0 = mb * 64 + wm * 32 + i * 16 + lh * 8;
      const int gn = nb * 64 + wn * 32 + j * 16 + lc;
      const int h = gn >> 6, d = gn & 63;
      const float biq = bq[gn], bik = bk[gn], biv = bv[gn];
#pragma unroll
      for (int e = 0; e < 8; ++e) {
        const int R = gm0 + e;
        const int b = R >> 11, s = R & (SS - 1);
        const size_t idx = ((size_t)((b * HH + h) * SS + s)) * DK + d;
        Q[idx] = acc[0][i][j][e] + biq;
        K[idx] = acc[1][i][j][e] + bik;
        V[idx] = acc[2][i][j][e] + biv;
      }
    }
}

// ---------------------------------------------------------------------------
// Kernel 2: masked attention logits.  logits = (Q K^T)/8, mask==0 -> -1e9.
// ---------------------------------------------------------------------------
__global__ __launch_bounds__(128) void scores_kernel(
    const float* __restrict__ Q, const float* __restrict__ K,
    const int* __restrict__ mask, float* __restrict__ attnW) {
  __shared__ float As[64 * LDT];
  __shared__ float Bs[64 * LDT];
  const int qb = blockIdx.x, kb = blockIdx.y, bh = blockIdx.z;
  const int b = bh >> 4;
  const float* Qb = Q + (size_t)bh * SS * DK;
  const float* Kb = K + (size_t)bh * SS * DK;

  stage64<LDT>(As, Qb + (size_t)(qb * 64) * DK, DK);
  stage64<LDT>(Bs, Kb + (size_t)(kb * 64) * DK, DK);
  wait_async<0>();
  __syncthreads();

  const int wave = threadIdx.x >> 5;
  const int wm = wave >> 1, wn = wave & 1;
  v8f zero = {};
  v8f acc[2][2] = {zero, zero, zero, zero};
  mma_tile(As, Bs, wm, wn, acc);

  const int lane = threadIdx.x & 31;
  const int lh = lane >> 4, lc = lane & 15;
#pragma unroll
  for (int i = 0; i < 2; ++i)
#pragma unroll
    for (int j = 0; j < 2; ++j) {
      const int gm0 = qb * 64 + wm * 32 + i * 16 + lh * 8;
      const int gn = kb * 64 + wn * 32 + j * 16 + lc;
      const int mk = mask[b * SS + gn];
#pragma unroll
      for (int e = 0; e < 8; ++e) {
        float val = acc[i][j][e] * 0.125f;  // 1/sqrt(64)
        if (mk == 0) val = -1e9f;
        attnW[((size_t)bh * SS + (gm0 + e)) * SS + gn] = val;
      }
    }
}

// ---------------------------------------------------------------------------
// Kernel 3: row softmax, in place, values register-resident:
// exactly one read + one write of the 536MB matrix.
// ---------------------------------------------------------------------------
__global__ __launch_bounds__(256) void softmax_kernel(float* __restrict__ w) {
  __shared__ float red[256];
  float4* p4 = reinterpret_cast<float4*>(w + (size_t)blockIdx.x * SS);
  const int t = threadIdx.x;
  float4 a = p4[t];
  float4 c = p4[t + 256];

  float m = fmaxf(fmaxf(fmaxf(a.x, a.y), fmaxf(a.z, a.w)),
                  fmaxf(fmaxf(c.x, c.y), fmaxf(c.z, c.w)));
  red[t] = m;
  __syncthreads();
  for (int s = 128; s > 0; s >>= 1) {
    if (t < s) red[t] = fmaxf(red[t], red[t + s]);
    __syncthreads();
  }
  const float rmax = red[0];
  __syncthreads();

  a.x = expf(a.x - rmax); a.y = expf(a.y - rmax);
  a.z = expf(a.z - rmax); a.w = expf(a.w - rmax);
  c.x = expf(c.x - rmax); c.y = expf(c.y - rmax);
  c.z = expf(c.z - rmax); c.w = expf(c.w - rmax);

  red[t] = a.x + a.y + a.z + a.w + c.x + c.y + c.z + c.w;
  __syncthreads();
  for (int s = 128; s > 0; s >>= 1) {
    if (t < s) red[t] += red[t + s];
    __syncthreads();
  }
  const float inv = 1.0f / red[0];

  a.x *= inv; a.y *= inv; a.z *= inv; a.w *= inv;
  c.x *= inv; c.y *= inv; c.z *= inv; c.w *= inv;
  p4[t] = a;
  p4[t + 256] = c;
}

// ---------------------------------------------------------------------------
// Kernel 4: context = attn_weights @ V per (b,h). Double-buffered async
// staging; V staged in natural [k][n] layout (stride LDV, conflict-free).
// ---------------------------------------------------------------------------
__global__ __launch_bounds__(128) void av_kernel(
    const float* __restrict__ attnW, const float* __restrict__ V,
    float* __restrict__ ctx) {
  __shared__ float As[2][64 * LDT];
  __shared__ float Vs[2][64 * LDV];
  const int qb = blockIdx.x, bh = blockIdx.z;
  const int b = bh >> 4, h = bh & 15;
  const float* Wb = attnW + (size_t)bh * SS * SS;
  const float* Vb = V + (size_t)bh * SS * DK;
  const int wave = threadIdx.x >> 5;
  const int wm = wave >> 1, wn = wave & 1;

  v8f zero = {};
  v8f acc[2][2] = {zero, zero, zero, zero};

  auto issue = [&](int buf, int k0) {
    stage64<LDT>(As[buf], Wb + (size_t)(qb * 64) * SS + k0, SS);
    stage64<LDV>(Vs[buf], Vb + (size_t)k0 * DK, DK);  // 16 issues/wave
  };

  issue(0, 0);
#pragma unroll 1
  for (int c = 0; c < SS / 64; ++c) {
    const int cur = c & 1;
    if (c + 1 < SS / 64) {
      issue(cur ^ 1, (c + 1) * 64);
      wait_async<16>();
    } else {
      wait_async<0>();
    }
    __syncthreads();
    mma_tile_nn(As[cur], Vs[cur], wm, wn, acc);
    __syncthreads();
  }

  const int lane = threadIdx.x & 31;
  const int lh = lane >> 4, lc = lane & 15;
#pragma unroll
  for (int i = 0; i < 2; ++i)
#pragma unroll
    for (int j = 0; j < 2; ++j) {
      const int gm0 = qb * 64 + wm * 32 + i * 16 + lh * 8;  // q position
      const int gn = wn * 32 + j * 16 + lc;                 // d within head
#pragma unroll
      for (int e = 0; e < 8; ++e)
        ctx[((size_t)(b * SS + gm0 + e)) * DD + h * DK + gn] = acc[i][j][e];
    }
}

// ---------------------------------------------------------------------------
// Kernel 5: output projection.  out = ctx @ wo^T + bo  (double-buffered async)
// ---------------------------------------------------------------------------
__global__ __launch_bounds__(128) void oproj_kernel(
    const float* __restrict__ ctx, const float* __restrict__ wo,
    const float* __restrict__ bo, float* __restrict__ out) {
  __shared__ float As[2][64 * LDT];
  __shared__ float Bs[2][64 * LDT];
  const int mb = blockIdx.x, nb = blockIdx.y;
  const int wave = threadIdx.x >> 5;
  const int wm = wave >> 1, wn = wave & 1;

  v8f zero = {};
  v8f acc[2][2] = {zero, zero, zero, zero};

  auto issue = [&](int buf, int k0) {
    stage64<LDT>(As[buf], ctx + (size_t)(mb * 64) * DD + k0, DD);
    stage64<LDT>(Bs[buf], wo + (size_t)(nb * 64) * DD + k0, DD);  // 16 issues/wave
  };

  issue(0, 0);
#pragma unroll 1
  for (int c = 0; c < DD / 64; ++c) {
    const int cur = c & 1;
    if (c + 1 < DD / 64) {
      issue(cur ^ 1, (c + 1) * 64);
      wait_async<16>();
    } else {
      wait_async<0>();
    }
    __syncthreads();
    mma_tile(As[cur], Bs[cur], wm, wn, acc);
    __syncthreads();
  }

  const int lane = threadIdx.x & 31;
  const int lh = lane >> 4, lc = lane & 15;
#pragma unroll
  for (int i = 0; i < 2; ++i)
#pragma unroll
    for (int j = 0; j < 2; ++j) {
      const int gm0 = mb * 64 + wm * 32 + i * 16 + lh * 8;
      const int gn = nb * 64 + wn * 32 + j * 16 + lc;
      const float bi = bo[gn];
#pragma unroll
      for (int e = 0; e < 8; ++e)
        out[(size_t)(gm0 + e) * DD + gn] = acc[i][j][e] + bi;
    }
}

// ---------------------------------------------------------------------------
extern "C" void kernel_launch(void* const* d_in, const int* in_sizes, int n_in,
                              void* d_out, int out_size, void* d_ws, size_t ws_size,
                              hipStream_t stream) {
  const float* x  = (const float*)d_in[0];
  const int* mask = (const int*)d_in[1];
  const float* wq = (const float*)d_in[2];
  const float* bq = (const float*)d_in[3];
  const float* wk = (const float*)d_in[4];
  const float* bk = (const float*)d_in[5];
  const float* wv = (const float*)d_in[6];
  const float* bv = (const float*)d_in[7];
  const float* wo = (const float*)d_in[8];
  const float* bo = (const float*)d_in[9];

  float* out   = (float*)d_out;                       // (B,S,D)
  float* attnW = out + (size_t)BB * SS * DD;          // (B,H,S,S)

  const size_t NQ = (size_t)BB * HH * SS * DK;        // 4,194,304 floats
  float* ws = (float*)d_ws;                           // needs 4*NQ*4 = 64 MB
  float* Q   = ws;
  float* K   = Q + NQ;
  float* V   = K + NQ;
  float* ctx = V + NQ;

  qkv_proj_kernel<<<dim3((BB * SS) / 64, DD / 64), 128, 0, stream>>>(
      x, wq, bq, wk, bk, wv, bv, Q, K, V);

  scores_kernel<<<dim3(SS / 64, SS / 64, BB * HH), 128, 0, stream>>>(
      Q, K, mask, attnW);

  softmax_kernel<<<dim3(BB * HH * SS), 256, 0, stream>>>(attnW);

  av_kernel<<<dim3(SS / 64, 1, BB * HH), 128, 0, stream>>>(attnW, V, ctx);

  oproj_kernel<<<dim3((BB * SS) / 64, DD / 64), 128, 0, stream>>>(
      ctx, wo, bo, out);
}